// SimpleCNNModel_62766652063749
// MI455X (gfx1250) — compile-verified
//
#include <hip/hip_runtime.h>

// ---------------------------------------------------------------------------
// LeNet-5 forward, fp32, B=8192, on gfx1250 (MI455X).
//   conv1+relu+buggy-pool  -> ws[0]   [B,6,14,14]   (A gathered from L2)
//   conv2+relu+buggy-pool  -> ws[1]   [B,400]       (A gathered from LDS tile)
//   fc1+fc2+fc3 (fused)    -> d_out   [B,10]
// All GEMM-shaped math uses V_WMMA_F32_16X16X4_F32 (fp32-exact matrix path).
// Hot loops are branchless: padded K rows of B are zeroed in LDS and padded N
// columns of D are never stored, so OOB is avoided by index clamping only.
// The stride-1 "buggy" 2x2 maxpool is an elementwise max of 4 accumulator
// fragments (C/D layout only depends on (image, cout)), done in registers.
// ---------------------------------------------------------------------------

typedef float v2f __attribute__((ext_vector_type(2)));
typedef float v8f __attribute__((ext_vector_type(8)));

__device__ __forceinline__ v8f wmma_f32_16x16x4(v2f a, v2f b, v8f c) {
  // (neg_a, A, neg_b, B, c_mod, C, reuse_a, reuse_b)
  return __builtin_amdgcn_wmma_f32_16x16x4_f32(false, a, false, b, (short)0, c,
                                               false, false);
}

__device__ __forceinline__ v8f zero8() {
  v8f z = {0.f, 0.f, 0.f, 0.f, 0.f, 0.f, 0.f, 0.f};
  return z;
}

__device__ __forceinline__ int imin(int a, int b) { return a < b ? a : b; }

// ---------------------------------------------------------------------------
// Fused conv(5x5,s1,VALID) + ReLU + buggy maxpool (2x2 window, stride 1,
// top-left PHxPW crop).  One workgroup = 16 images (M tile).  Each wave owns
// pool rows r, r+nwaves, ...: computes conv rows r,r+1 streaming over x and
// pools in registers.  LDSIN: stage the touched (PH+5)x(PW+5) input subregion
// per image in LDS (odd stride -> conflict-free) and gather A via ds_load.
// ---------------------------------------------------------------------------
template <int CIN, int HIN, int WIN, int COUT, int PH, int PW, bool LDSIN>
__global__ void conv_relu_pool(const float* __restrict__ in,
                               const float* __restrict__ w,
                               float* __restrict__ out) {
  constexpr int K    = CIN * 25;      // GEMM reduction dim
  constexpr int NC   = (K + 3) / 4;   // K chunks of 4
  constexpr int KP   = NC * 4;        // padded K (even -> b64-aligned rows)
  constexpr int IMG  = CIN * HIN * WIN;
  constexpr int SUBH = PH + 5, SUBW = PW + 5;     // touched input subregion
  constexpr int ISZ  = CIN * SUBH * SUBW;
  constexpr int IMST = ISZ | 1;                   // odd per-image LDS stride

  __shared__ float wlds[16 * KP];                 // B matrix [n][k], zero-pad
  __shared__ float ilds[LDSIN ? 16 * IMST : 1];   // staged input subregions

  const int tid = threadIdx.x;
  const int b0  = blockIdx.x * 16;

  for (int i = tid; i < 16 * KP; i += blockDim.x) {
    const int n = i / KP, k = i % KP;
    wlds[i] = (n < COUT && k < K) ? w[n * K + k] : 0.f;
  }
  if (LDSIN) {
    for (int i = tid; i < 16 * ISZ; i += blockDim.x) {
      const int img = i / ISZ, rem = i % ISZ;
      const int ci = rem / (SUBH * SUBW), rc = rem % (SUBH * SUBW);
      const int row = rc / SUBW, col = rc % SUBW;
      ilds[img * IMST + rem] =
          in[(size_t)(b0 + img) * IMG + ci * (HIN * WIN) + row * WIN + col];
    }
  }
  __syncthreads();

  const int lane   = tid & 31;
  const int wave   = tid >> 5;
  const int nwaves = blockDim.x >> 5;
  const int m      = lane & 15;   // image within tile (A row / D col sel)
  const int kh     = lane >> 4;   // K half selector

  // Hoist all B fragments into registers (loop-invariant over (r,x)).
  v2f bfr[NC];
#pragma unroll
  for (int kc = 0; kc < NC; ++kc)
    bfr[kc] = *(const v2f*)&wlds[m * KP + 4 * kc + 2 * kh];

  const float* imgbase = in + (size_t)(b0 + m) * IMG;  // (!LDSIN path)
  const int    ldsbase = m * IMST;                     // (LDSIN path)

  for (int r = wave; r < PH; r += nwaves) {       // pool row
    v8f mprev = zero8();
    for (int x = 0; x < PW + 1; ++x) {            // conv column
      v8f acc[2];
#pragma unroll
      for (int half = 0; half < 2; ++half) {      // conv rows r, r+1
        const int y = r + half;
        v8f c = zero8();
#pragma unroll
        for (int kc = 0; kc < NC; ++kc) {
          const int k0 = 4 * kc + 2 * kh;
          v2f af;
#pragma unroll
          for (int v = 0; v < 2; ++v) {
            // Branchless: clamp padded k to a valid address; the value is
            // multiplied by a zeroed B row, so it never reaches D.
            const int kk = imin(k0 + v, K - 1);
            const int ci = kk / 25, rr = kk % 25;
            const int ky = rr / 5, kx = rr % 5;
            af[v] = LDSIN
                ? ilds[ldsbase + ci * (SUBH * SUBW) + (y + ky) * SUBW + (x + kx)]
                : imgbase[ci * (HIN * WIN) + (y + ky) * WIN + (x + kx)];
          }
          c = wmma_f32_16x16x4(af, bfr[kc], c);
        }
        acc[half] = c;
      }
      // relu + vertical max, all in the (m, n) slot of the same lane/vgpr
      v8f mx;
#pragma unroll
      for (int i = 0; i < 8; ++i)
        mx[i] = fmaxf(fmaxf(acc[0][i], 0.f), fmaxf(acc[1][i], 0.f));
      if (x > 0) {
        const int n = lane & 15;   // cout
        if (n < COUT) {
#pragma unroll
          for (int v = 0; v < 8; ++v) {
            const int mm = v + 8 * kh;              // image row of D element v
            const float pv = fmaxf(mprev[v], mx[v]);
            out[((size_t)(b0 + mm) * COUT + n) * (PH * PW) + r * PW + (x - 1)] = pv;
          }
        }
      }
      mprev = mx;
    }
  }
}

// ---------------------------------------------------------------------------
// Fused FC stack: relu(in*W1^T+b1) -> relu(*W2^T+b2) -> *W3^T+b3.
// One workgroup = 16 batch rows, 8 waves.  Activations ping-pong through LDS
// with even strides (8B-aligned b64, good bank spread).  B rows are fetched
// with a clamped row index: padded N columns of D are never stored, so the
// inner loops carry no divergence at all.
// ---------------------------------------------------------------------------
__global__ void fc_fused(const float* __restrict__ in,   // [B,400]
                         const float* __restrict__ w1,   // [120,400]
                         const float* __restrict__ b1,   // [120]
                         const float* __restrict__ w2,   // [84,120]
                         const float* __restrict__ b2,   // [84]
                         const float* __restrict__ w3,   // [10,84]
                         const float* __restrict__ b3,   // [10]
                         float* __restrict__ out) {      // [B,10]
  constexpr int ST0 = 402, ST1 = 122, ST2 = 86;
  __shared__ float a0[16 * ST0];
  __shared__ float a1[16 * ST1];
  __shared__ float a2[16 * ST2];

  const int tid = threadIdx.x;
  const int b0  = blockIdx.x * 16;

  // stage input tile [16][400] via b64
  for (int i = tid; i < 16 * 200; i += blockDim.x) {
    const int mr = i / 200, j = i % 200;
    const v2f val = *(const v2f*)(in + (size_t)(b0 + mr) * 400 + 2 * j);
    *(v2f*)&a0[mr * ST0 + 2 * j] = val;
  }
  __syncthreads();

  const int lane = tid & 31;
  const int wave = tid >> 5;
  const int n16  = lane & 15;
  const int kh   = lane >> 4;

  // ---- fc1: N=120 (8 tiles), K=400 (100 chunks) ----
  {
    const int n = wave * 16 + n16;
    const float* wrow = w1 + (size_t)imin(n, 119) * 400;  // clamped, branchless
    v8f c = zero8();
    for (int kc = 0; kc < 100; ++kc) {
      const int k0 = 4 * kc + 2 * kh;
      const v2f av = *(const v2f*)&a0[n16 * ST0 + k0];
      const v2f bv = *(const v2f*)(wrow + k0);
      c = wmma_f32_16x16x4(av, bv, c);
    }
    if (n < 120) {
      const float bias = b1[n];
#pragma unroll
      for (int v = 0; v < 8; ++v)
        a1[(v + 8 * kh) * ST1 + n] = fmaxf(c[v] + bias, 0.f);
    }
  }
  __syncthreads();

  // ---- fc2: N=84 (6 tiles), K=120 (30 chunks) ----
  if (wave < 6) {
    const int n = wave * 16 + n16;
    const float* wrow = w2 + (size_t)imin(n, 83) * 120;
    v8f c = zero8();
    for (int kc = 0; kc < 30; ++kc) {
      const int k0 = 4 * kc + 2 * kh;
      const v2f av = *(const v2f*)&a1[n16 * ST1 + k0];
      const v2f bv = *(const v2f*)(wrow + k0);
      c = wmma_f32_16x16x4(av, bv, c);
    }
    if (n < 84) {
      const float bias = b2[n];
#pragma unroll
      for (int v = 0; v < 8; ++v)
        a2[(v + 8 * kh) * ST2 + n] = fmaxf(c[v] + bias, 0.f);
    }
  }
  __syncthreads();

  // ---- fc3: N=10 (1 tile), K=84 (21 chunks) ----
  if (wave == 0) {
    const int n = n16;
    const float* wrow = w3 + (size_t)imin(n, 9) * 84;
    v8f c = zero8();
#pragma unroll
    for (int kc = 0; kc < 21; ++kc) {
      const int k0 = 4 * kc + 2 * kh;
      const v2f av = *(const v2f*)&a2[n16 * ST2 + k0];
      const v2f bv = *(const v2f*)(wrow + k0);
      c = wmma_f32_16x16x4(av, bv, c);
    }
    if (n < 10) {
      const float bias = b3[n];
#pragma unroll
      for (int v = 0; v < 8; ++v)
        out[(size_t)(b0 + v + 8 * kh) * 10 + n] = c[v] + bias;
    }
  }
}

// ---------------------------------------------------------------------------
extern "C" void kernel_launch(void* const* d_in, const int* in_sizes, int n_in,
                              void* d_out, int out_size, void* d_ws, size_t ws_size,
                              hipStream_t stream) {
  const float* x   = (const float*)d_in[0];  // [B,3,32,32]
  const float* c1w = (const float*)d_in[1];  // [6,3,5,5]
  const float* c2w = (const float*)d_in[2];  // [16,6,5,5]
  const float* f1w = (const float*)d_in[3];  // [120,400]
  const float* f1b = (const float*)d_in[4];
  const float* f2w = (const float*)d_in[5];  // [84,120]
  const float* f2b = (const float*)d_in[6];
  const float* f3w = (const float*)d_in[7];  // [10,84]
  const float* f3b = (const float*)d_in[8];
  float* outp = (float*)d_out;

  const int B = in_sizes[0] / (3 * 32 * 32);  // 8192

  float* out1 = (float*)d_ws;                   // [B,6,14,14]
  float* out2 = out1 + (size_t)B * 6 * 14 * 14; // [B,400]

  const int ntiles = B / 16;

  // conv1: CIN=3,32x32 -> pool 14x14, COUT=6. 8 waves/WG, global A gather.
  conv_relu_pool<3, 32, 32, 6, 14, 14, false>
      <<<ntiles, 256, 0, stream>>>(x, c1w, out1);
  // conv2: CIN=6,14x14 -> pool 5x5, COUT=16. 5 waves/WG, LDS-staged A gather.
  conv_relu_pool<6, 14, 14, 16, 5, 5, true>
      <<<ntiles, 160, 0, stream>>>(out1, c2w, out2);
  // fused FC stack.
  fc_fused<<<ntiles, 256, 0, stream>>>(out2, f1w, f1b, f2w, f2b, f3w, f3b, outp);
}